// DegradedBicycleRollout_80607946212043
// MI455X (gfx1250) — compile-verified
//
#include <hip/hip_runtime.h>

// DegradedBicycleRollout for MI455X (gfx1250, wave32).
// One thread per (b,l) trajectory; sequential 80-step scan in registers.
// Bandwidth-bound (~159 MB -> ~6.8 us floor @ 23.3 TB/s) AND chain-bound
// (only 1024 waves total). Key optimizations:
//   - loop-carried state reduced to {px,py,psi,speed_sq}; speed_sq updated
//     as speed2^2 (== vx2^2+vy2^2 exactly), so sincos/tan/atan/exp are all
//     OFF the recurrence chain and parallelize across unrolled steps.
//   - small-range polynomial atan (arg bounded to |x|<=0.419 by construction)
//   - b128 non-temporal stores/loads + global_prefetch_b8 on the ctrl stream

typedef float f4 __attribute__((ext_vector_type(4)));

#define DT_F     0.1f
#define INV_DT   10.0f
#define INV_WB   (1.0f / 2.8f)

__device__ __forceinline__ float fast_rcp(float x) {
    return __builtin_amdgcn_rcpf(x);          // v_rcp_f32
}
__device__ __forceinline__ float fast_sigmoid(float x) {
    return fast_rcp(1.0f + __expf(-x));       // v_exp_f32 + v_rcp_f32
}
__device__ __forceinline__ float fast_tanh(float x) {
    // tanh(x) = 2*sigmoid(2x) - 1  -> one exp + one rcp
    return 2.0f * fast_rcp(1.0f + __expf(-2.0f * x)) - 1.0f;
}
__device__ __forceinline__ float clampf(float x, float lo, float hi) {
    return fminf(fmaxf(x, lo), hi);
}
// atan for |x| <= ~0.42 (Cephes small-range odd polynomial, no reduction).
__device__ __forceinline__ float atan_small(float x) {
    const float x2 = x * x;
    float p = 8.05374449538e-2f;
    p = __builtin_fmaf(p, x2, -1.38776856032e-1f);
    p = __builtin_fmaf(p, x2,  1.99777106478e-1f);
    p = __builtin_fmaf(p, x2, -3.33329491539e-1f);
    return __builtin_fmaf(p * x2, x, x);
}
// tan for |x| <= 0.75 via hardware sin/cos + rcp (off the recurrence chain).
__device__ __forceinline__ float tan_small(float x) {
    return __sinf(x) * fast_rcp(__cosf(x));
}

template <int L, int H>
__global__ __launch_bounds__(128)
void degraded_bicycle_rollout_kernel(const float* __restrict__ x0,
                                     const float* __restrict__ ctrl,
                                     const float* __restrict__ deg,
                                     float* __restrict__ out,
                                     int total) {
    const int tid = blockIdx.x * blockDim.x + threadIdx.x;
    if (tid >= total) return;

    const int b = tid / L;   // L = 64 -> shift

    const float* dg = deg + (size_t)b * 5;
    const float steer = fmaxf(dg[0], 0.05f);
    const float brake = fmaxf(dg[1], 0.05f);
    const float thr   = fmaxf(dg[2], 0.05f);
    const float fric  = fmaxf(dg[4], 0.10f);

    // Per-trajectory bases: ctrl run 960 B, out run 3888 B, both 16B-aligned.
    const f4* __restrict__ cp = (const f4*)(ctrl + (size_t)tid * (H * 3));
    f4* __restrict__ op = (f4*)(out + (size_t)tid * ((H + 1) * 12));

    // out[..., 0, :] = broadcast x0[b, :]  (48 B, 16B-aligned)
    const f4* __restrict__ xp = (const f4*)(x0 + (size_t)b * 12);
    f4 s0 = xp[0];
    f4 s1 = xp[1];
    f4 s2 = xp[2];
    __builtin_nontemporal_store(s0, op + 0);
    __builtin_nontemporal_store(s1, op + 1);
    __builtin_nontemporal_store(s2, op + 2);

    float px = s0.x, py = s0.y, psi = s0.z, vx = s0.w, vy = s1.x;
    float speed_sq = vx * vx + vy * vy;   // loop-carried instead of (vx,vy)

    const float acc_floor  = -7.5f * fric;
    const float yawlim_num = fric * 9.81f;

#pragma unroll 2
    for (int ch = 0; ch < H / 4; ++ch) {
        f4 ca = __builtin_nontemporal_load(cp + 3 * ch + 0);
        f4 cb = __builtin_nontemporal_load(cp + 3 * ch + 1);
        f4 cc = __builtin_nontemporal_load(cp + 3 * ch + 2);

        // Prefetch the control chunk two iterations ahead (global_prefetch_b8).
        if (ch + 2 < H / 4) {
            __builtin_prefetch((const void*)(cp + 3 * ch + 6), 0, 0);
        }

        const float u[12] = {ca.x, ca.y, ca.z, ca.w,
                             cb.x, cb.y, cb.z, cb.w,
                             cc.x, cc.y, cc.z, cc.w};

#pragma unroll
        for (int j = 0; j < 4; ++j) {
            const int t = ch * 4 + j;

            // ---- control-only computations: OFF the recurrence chain ----
            const float delta = steer * fast_tanh(u[3 * j + 0]);
            const float fb    = brake * fast_sigmoid(u[3 * j + 1]);
            const float fx    = thr   * fast_sigmoid(u[3 * j + 2]);
            const float acc   = fminf(fmaxf(2.8f * fx - 6.5f * fb, acc_floor), 3.0f);
            const float tan_d = tan_small(clampf(delta, -0.75f, 0.75f));
            // |0.45*tan_d| <= 0.419 -> small-range atan; +-0.65 clip kept.
            const float beta  = clampf(atan_small(0.45f * tan_d), -0.65f, 0.65f);

            // ---- short recurrence: speed_sq -> sqrt -> speed2 ----
            const float speed  = __builtin_amdgcn_sqrtf(speed_sq + 1e-6f);
            const float speed2 = fmaxf(__builtin_fmaf(acc, DT_F, speed), 0.0f);

            const float raw_yaw  = speed2 * INV_WB * tan_d;
            const float yaw_lim  = fmaxf(yawlim_num * fast_rcp(fmaxf(speed2, 2.0f)), 0.15f);
            const float yaw_rate = clampf(raw_yaw, -1.0f, 1.0f) * yaw_lim;
            const float psi2     = __builtin_fmaf(yaw_rate, DT_F, psi);

            // ---- output-only branch (sincos hangs off the chain) ----
            float sn, cs;
            __sincosf(psi2 + beta, &sn, &cs);
            const float vx2 = speed2 * cs;
            const float vy2 = speed2 * sn;
            const float px2 = __builtin_fmaf(vx2, DT_F, px);
            const float py2 = __builtin_fmaf(vy2, DT_F, py);
            const float ax = (vx2 - vx) * INV_DT;
            const float ay = (vy2 - vy) * INV_DT;

            f4 o0 = {px2, py2, psi2, vx2};
            f4 o1 = {vy2, yaw_rate, ax, ay};
            f4 o2 = {beta, delta, fb, fx};
            f4* od = op + (size_t)(t + 1) * 3;
            __builtin_nontemporal_store(o0, od + 0);
            __builtin_nontemporal_store(o1, od + 1);
            __builtin_nontemporal_store(o2, od + 2);

            // carry state: vx2^2 + vy2^2 == speed2^2 (sin^2+cos^2 == 1)
            px = px2; py = py2; psi = psi2;
            vx = vx2; vy = vy2;
            speed_sq = speed2 * speed2;
        }
    }
}

extern "C" void kernel_launch(void* const* d_in, const int* in_sizes, int n_in,
                              void* d_out, int out_size, void* d_ws, size_t ws_size,
                              hipStream_t stream) {
    (void)n_in; (void)out_size; (void)d_ws; (void)ws_size;

    const float* x0   = (const float*)d_in[0];   // [B, 12]
    const float* ctrl = (const float*)d_in[1];   // [B, L, H, 3]
    const float* deg  = (const float*)d_in[2];   // [B, 5]
    float* out = (float*)d_out;                  // [B, L, H+1, 12]

    constexpr int L = 64;
    constexpr int H = 80;
    const int B = in_sizes[0] / 12;
    const int total = B * L;                     // 32768 threads = 1024 wave32

    const dim3 block(128);                       // 4 waves/block, fine-grain spread
    const dim3 grid((total + 127) / 128);
    hipLaunchKernelGGL((degraded_bicycle_rollout_kernel<L, H>), grid, block, 0, stream,
                       x0, ctrl, deg, out, total);
}